// NodeFeat_61512521613940
// MI455X (gfx1250) — compile-verified
//
#include <hip/hip_runtime.h>
#include <hip/hip_bf16.h>
#include <stdint.h>

// ---------------------------------------------------------------------------
// Layout: workspace H is [N, 576] row-major.
//   H[n,   0..191] = x3  = [x, x*rsqrt(deg), x*sqrt(deg)]
//   H[n, 192..383] = xs1 = degrev * spmm(x3)
//   H[n, 384..575] = xs2 = degrev * spmm(xs1) - x3
// Output: out[s,e,d,j] = H[edge[s,e], j*64 + d],  flat = idx*576 + d*9 + j
// ---------------------------------------------------------------------------

#define DFEAT 64
#define ROW   576            // 3*(HOP+1)*DFEAT
#define FPL   6              // 192 feature lanes / 32 = 6 floats per lane
#define EDGES_PER_WAVE 16
#define GATHER_WAVES 8       // waves per block in gather kernel
#define JBLK  68             // padded 64-float j-block (bank-conflict-free reads)
#define LROW  (9 * JBLK)     // padded LDS row: 612 floats

#if defined(__has_builtin)
#  if __has_builtin(__builtin_amdgcn_global_load_async_to_lds_b64)
#    define HAVE_ASYNC_LDS 1
#  endif
#endif
#ifndef HAVE_ASYNC_LDS
#  define HAVE_ASYNC_LDS 0
#endif

#if HAVE_ASYNC_LDS
typedef int v2i __attribute__((vector_size(8)));
typedef __attribute__((address_space(1))) v2i* g_v2i_p;   // global (b64 payload)
typedef __attribute__((address_space(3))) v2i* l_v2i_p;   // LDS    (b64 payload)

// Generic -> AS1: identity bits for global pointers.
__device__ __forceinline__ g_v2i_p as_global_v2i(const void* p) {
  return (g_v2i_p)(unsigned long long)(uintptr_t)p;
}
// Generic -> AS3: flat LDS address truncates to the 32-bit LDS offset
__device__ __forceinline__ l_v2i_p as_lds_v2i(void* p) {
  return (l_v2i_p)(unsigned int)(uintptr_t)p;
}
#endif

__device__ __forceinline__ void wait_async_zero() {
#if defined(__has_builtin) && __has_builtin(__builtin_amdgcn_s_wait_asynccnt)
  __builtin_amdgcn_s_wait_asynccnt(0);
#else
  asm volatile("s_wait_asynccnt 0" ::: "memory");
#endif
}

// --- K1: build x3 and zero both hop destination regions ---------------------
__global__ void build_x3_kernel(const float* __restrict__ x,
                                const float* __restrict__ deg,
                                float* __restrict__ H, int N) {
  int tid = blockIdx.x * blockDim.x + threadIdx.x;
  if (tid >= N * DFEAT) return;
  int n = tid >> 6;
  int d = tid & 63;
  float v  = x[(size_t)n * DFEAT + d];
  float dg = deg[n];
  float rs = rsqrtf(dg);
  float sq = sqrtf(dg);
  size_t rb = (size_t)n * ROW;
  H[rb + d]        = v;
  H[rb + 64 + d]   = v * rs;
  H[rb + 128 + d]  = v * sq;
#pragma unroll
  for (int k = 0; k < 6; ++k) H[rb + 192 + k * 64 + d] = 0.0f;
}

// --- K2/K4: SpMM y[row] += src[col] exploiting sorted rows ------------------
// One wave32 per EDGES_PER_WAVE consecutive edges; lane owns features
// {lane, lane+32, ..., lane+160}. Accumulate in registers while the
// (wave-uniform) row is unchanged; flush with f32 atomics on row change.
__global__ void spmm_kernel(float* __restrict__ H,
                            const int* __restrict__ adj_row,
                            const int* __restrict__ adj_col,
                            int srcOff, int dstOff, int nnz) {
  int lane = threadIdx.x & 31;
  int wave = blockIdx.x * (blockDim.x >> 5) + (threadIdx.x >> 5);
  int e0 = wave * EDGES_PER_WAVE;
  if (e0 >= nnz) return;
  int e1 = e0 + EDGES_PER_WAVE;
  if (e1 > nnz) e1 = nnz;

  float acc[FPL];
#pragma unroll
  for (int i = 0; i < FPL; ++i) acc[i] = 0.0f;

  int curRow = adj_row[e0];
  for (int e = e0; e < e1; ++e) {
    int r = adj_row[e];
    int c = adj_col[e];
    if (e + 1 < e1) {  // gfx1250 prefetch of next random source row
      __builtin_prefetch(&H[(size_t)adj_col[e + 1] * ROW + srcOff], 0, 0);
    }
    if (r != curRow) {  // wave-uniform branch
      float* dst = H + (size_t)curRow * ROW + dstOff + lane;
#pragma unroll
      for (int i = 0; i < FPL; ++i) {
        atomicAdd(dst + i * 32, acc[i]);
        acc[i] = 0.0f;
      }
      curRow = r;
    }
    const float* src = H + (size_t)c * ROW + srcOff + lane;
#pragma unroll
    for (int i = 0; i < FPL; ++i) acc[i] += src[i * 32];
  }
  float* dst = H + (size_t)curRow * ROW + dstOff + lane;
#pragma unroll
  for (int i = 0; i < FPL; ++i) atomicAdd(dst + i * 32, acc[i]);
}

// --- K3/K5: scale by 1/deg, optional Chebyshev subtract ---------------------
__global__ void finish_kernel(float* __restrict__ H,
                              const float* __restrict__ deg,
                              int dstOff, int subOff, int N) {
  int tid = blockIdx.x * blockDim.x + threadIdx.x;
  if (tid >= N * 192) return;
  int n = tid / 192;
  int d = tid - n * 192;
  size_t rb = (size_t)n * ROW;
  float v = H[rb + dstOff + d] * (1.0f / deg[n]);
  if (subOff >= 0) v -= H[rb + subOff + d];
  H[rb + dstOff + d] = v;
}

// --- K6: gather edge endpoints + per-node [9,64]->[64,9] transpose ----------
// One wave per edge. Async-DMA the 2304B node row into this wave's LDS slice:
// each of the 9 b64-per-lane async copies moves one 64-float j-block into a
// 68-float padded slot, so the transposed reads stage[j*68+d] hit bank
// (4j+d) mod 64 -> conflict-free within every aligned 32-lane group. Then
// s_wait_asynccnt and emit 18 fully coalesced 128B transposed stores.
__global__ void gather_kernel(const float* __restrict__ H,
                              const int* __restrict__ edge,
                              float* __restrict__ out, int nEdgesTotal) {
  __shared__ float stage[GATHER_WAVES][LROW];
  int lane = threadIdx.x & 31;
  int wave = threadIdx.x >> 5;
  int idx = blockIdx.x * GATHER_WAVES + wave;
  if (idx >= nEdgesTotal) return;

  int node = edge[idx];
  const float* row = H + (size_t)node * ROW;
  float* dst = stage[wave];
  size_t ob = (size_t)idx * ROW;

#if HAVE_ASYNC_LDS
#pragma unroll
  for (int jb = 0; jb < 9; ++jb) {
    g_v2i_p gp = as_global_v2i(row + jb * 64 + (lane << 1));
    l_v2i_p lp = as_lds_v2i(dst + jb * JBLK + (lane << 1));
    __builtin_amdgcn_global_load_async_to_lds_b64(gp, lp, 0, 0);
  }
  wait_async_zero();
#pragma unroll
  for (int k = 0; k < 18; ++k) {
    int o = k * 32 + lane;     // output index within row: o = d*9 + j
    int d = o / 9;
    int j = o - d * 9;
    out[ob + o] = dst[j * JBLK + d];
  }
#else
  __builtin_prefetch(row, 0, 0);
#pragma unroll
  for (int k = 0; k < 18; ++k) {
    int o = k * 32 + lane;
    int d = o / 9;
    int j = o - d * 9;
    out[ob + o] = row[j * 64 + d];
  }
  (void)dst;
#endif
}

// ---------------------------------------------------------------------------
extern "C" void kernel_launch(void* const* d_in, const int* in_sizes, int n_in,
                              void* d_out, int out_size, void* d_ws, size_t ws_size,
                              hipStream_t stream) {
  (void)n_in; (void)out_size; (void)ws_size;
  const float* x       = (const float*)d_in[0];
  const float* deg     = (const float*)d_in[1];
  const int*   adj_row = (const int*)d_in[2];
  const int*   adj_col = (const int*)d_in[3];
  const int*   edge    = (const int*)d_in[4];
  float* out = (float*)d_out;
  float* H   = (float*)d_ws;   // [N, 576] feature buffer

  const int N   = in_sizes[1];        // deg is [N,1]
  const int nnz = in_sizes[2];
  const int nq  = in_sizes[4];        // 2 * N_QEDGES flat

  // K1: x3 + zero hop regions
  {
    int th = N * DFEAT;
    build_x3_kernel<<<(th + 255) / 256, 256, 0, stream>>>(x, deg, H, N);
  }
  // K2: y1 = spmm(x3)
  {
    int waves  = (nnz + EDGES_PER_WAVE - 1) / EDGES_PER_WAVE;
    int blocks = (waves + 7) / 8;
    spmm_kernel<<<blocks, 256, 0, stream>>>(H, adj_row, adj_col, 0, 192, nnz);
  }
  // K3: xs1 = degrev * y1
  {
    int th = N * 192;
    finish_kernel<<<(th + 255) / 256, 256, 0, stream>>>(H, deg, 192, -1, N);
  }
  // K4: y2 = spmm(xs1)
  {
    int waves  = (nnz + EDGES_PER_WAVE - 1) / EDGES_PER_WAVE;
    int blocks = (waves + 7) / 8;
    spmm_kernel<<<blocks, 256, 0, stream>>>(H, adj_row, adj_col, 192, 384, nnz);
  }
  // K5: xs2 = degrev * y2 - x3
  {
    int th = N * 192;
    finish_kernel<<<(th + 255) / 256, 256, 0, stream>>>(H, deg, 384, 0, N);
  }
  // K6: out = transpose(H)[edge]
  {
    int blocks = (nq + GATHER_WAVES - 1) / GATHER_WAVES;
    gather_kernel<<<blocks, 256, 0, stream>>>(H, edge, out, nq);
  }
}